// GraphMambaBlock_8546984919675
// MI455X (gfx1250) — compile-verified
//
#include <hip/hip_runtime.h>
#include <math.h>

// ---------------- problem constants ----------------
constexpr int NN = 32768;      // nodes / sequence length
constexpr int DD = 128;        // model dim
constexpr int EE = 524288;     // edges
constexpr int DSTATE = 16;
constexpr int DTRANK = 8;
constexpr int PROJW = DTRANK + 2 * DSTATE;   // 40
constexpr int CH = 256;        // scan chunk length
constexpr int NCH = NN / CH;   // 128 chunks
constexpr float LN_EPS = 1e-5f;
constexpr float BN_EPS = 1e-5f;

typedef float v2f __attribute__((ext_vector_type(2)));
typedef float v8f __attribute__((ext_vector_type(8)));

__device__ __forceinline__ float siluf(float v) { return v / (1.f + expf(-v)); }

// ================= WMMA GEMM (fp32, 16x16x4) =================
// C[M,Nout] = act( A[M,K](row-major, lda) @ W[K,Nout] + bias (+ addsrc) )
// optional row gather on A (arow) and row scatter on C (orow).
// act: 0=none 1=relu 2=gelu(exact) 3=softplus
// Out-of-range columns (Nout not multiple of 16) are handled by CLAMPING the
// load address: column n of B only ever contributes to column n of D, so
// garbage in clamped columns never contaminates stored output; the store is
// guarded. This keeps the inner loop free of EXEC manipulation so the
// load -> v_wmma pipeline stays tight.
__global__ void gemm_wmma_f32(const float* __restrict__ A, int lda,
                              const float* __restrict__ W,
                              const float* __restrict__ bias,
                              const float* __restrict__ addsrc,
                              const int* __restrict__ arow,
                              const int* __restrict__ orow,
                              float* __restrict__ C,
                              int M, int K, int Nout, int act) {
  const int wv = threadIdx.x >> 5;
  const int lane = threadIdx.x & 31;
  const int tilesN = (Nout + 15) >> 4;
  const int tid = blockIdx.x * 4 + wv;
  const int totalTiles = (M >> 4) * tilesN;
  if (tid >= totalTiles) return;                 // wave-uniform -> EXEC stays full
  const int tm = tid / tilesN;
  const int tn = tid - tm * tilesN;
  const int mrow = tm * 16 + (lane & 15);
  const int ar = arow ? arow[mrow] : mrow;
  const int col = tn * 16 + (lane & 15);
  const int kh = lane >> 4;
  const bool colok = col < Nout;
  const int colc = colok ? col : (Nout - 1);     // clamped address, branch-free loads
  const float* Ap = A + (size_t)ar * lda;
  const float* Wp = W + colc;
  v8f acc = {0.f, 0.f, 0.f, 0.f, 0.f, 0.f, 0.f, 0.f};
  for (int k0 = 0; k0 < K; k0 += 4) {
    const int ka = k0 + 2 * kh;
    v2f a, b;
    a.x = Ap[ka];
    a.y = Ap[ka + 1];
    b.x = Wp[(size_t)ka * Nout];
    b.y = Wp[(size_t)(ka + 1) * Nout];
    acc = __builtin_amdgcn_wmma_f32_16x16x4_f32(false, a, false, b,
                                                (short)0, acc, false, false);
  }
  if (!colok) return;
  const float bv = bias ? bias[col] : 0.f;
#pragma unroll
  for (int v = 0; v < 8; v++) {
    const int r = tm * 16 + v + 8 * kh;
    float val = acc[v] + bv;
    if (addsrc) val += addsrc[(size_t)r * Nout + col];
    if (act == 1) val = val > 0.f ? val : 0.f;
    else if (act == 2) val = 0.5f * val * (1.f + erff(val * 0.70710678118654752440f));
    else if (act == 3) val = (val > 20.f) ? val : log1pf(expf(val));
    const int ro = orow ? orow[r] : r;
    C[(size_t)ro * Nout + col] = val;
  }
}

// ================= row layernorm over D=128 =================
__global__ void row_ln_kernel(const float* __restrict__ in,
                              const float* __restrict__ w,
                              const float* __restrict__ b,
                              float* __restrict__ out) {
  __shared__ float sh[128];
  const int r = blockIdx.x, c = threadIdx.x;
  const float v = in[(size_t)r * 128 + c];
  sh[c] = v; __syncthreads();
  for (int o = 64; o > 0; o >>= 1) { if (c < o) sh[c] += sh[c + o]; __syncthreads(); }
  const float mu = sh[0] * (1.f / 128.f);
  __syncthreads();
  const float dv = v - mu;
  sh[c] = dv * dv; __syncthreads();
  for (int o = 64; o > 0; o >>= 1) { if (c < o) sh[c] += sh[c + o]; __syncthreads(); }
  const float istd = rsqrtf(sh[0] * (1.f / 128.f) + LN_EPS);
  out[(size_t)r * 128 + c] = dv * istd * w[c] + b[c];
}

// fused batchnorm (per column) then row layernorm
__global__ void bn_ln_kernel(const float* __restrict__ in,
                             const float* __restrict__ bnmu,
                             const float* __restrict__ bnistd,
                             const float* __restrict__ bnw,
                             const float* __restrict__ bnb,
                             const float* __restrict__ fnw,
                             const float* __restrict__ fnb,
                             float* __restrict__ out) {
  __shared__ float sh[128];
  const int r = blockIdx.x, c = threadIdx.x;
  float v = in[(size_t)r * 128 + c];
  v = (v - bnmu[c]) * bnistd[c] * bnw[c] + bnb[c];
  sh[c] = v; __syncthreads();
  for (int o = 64; o > 0; o >>= 1) { if (c < o) sh[c] += sh[c + o]; __syncthreads(); }
  const float mu = sh[0] * (1.f / 128.f);
  __syncthreads();
  const float dv = v - mu;
  sh[c] = dv * dv; __syncthreads();
  for (int o = 64; o > 0; o >>= 1) { if (c < o) sh[c] += sh[c + o]; __syncthreads(); }
  const float istd = rsqrtf(sh[0] * (1.f / 128.f) + LN_EPS);
  out[(size_t)r * 128 + c] = dv * istd * fnw[c] + fnb[c];
}

// ================= graph degree / CSR =================
__global__ void degree_kernel(const int* __restrict__ src, const int* __restrict__ dst,
                              int* __restrict__ degS, int* __restrict__ degI, int e) {
  const int i = blockIdx.x * blockDim.x + threadIdx.x;
  if (i >= e) return;
  atomicAdd(&degS[src[i]], 1);
  atomicAdd(&degI[dst[i]], 1);
}

// exclusive prefix over 32768 ints; one block of 1024 threads, 32 each
__global__ void prefix_kernel(const int* __restrict__ deg, int* __restrict__ offs, int n) {
  __shared__ int sh[1024];
  const int t = threadIdx.x;
  const int base = t * 32;
  int s = 0;
  for (int i = 0; i < 32; i++) s += deg[base + i];
  sh[t] = s; __syncthreads();
  for (int o = 1; o < 1024; o <<= 1) {
    int val = (t >= o) ? sh[t - o] : 0;
    __syncthreads();
    sh[t] += val;
    __syncthreads();
  }
  int run = sh[t] - s;   // exclusive base for this thread's range
  for (int i = 0; i < 32; i++) { offs[base + i] = run; run += deg[base + i]; }
  if (t == 1023) offs[n] = sh[1023];
}

__global__ void copy_int_kernel(const int* __restrict__ a, int* __restrict__ b, int n) {
  const int i = blockIdx.x * blockDim.x + threadIdx.x;
  if (i < n) b[i] = a[i];
}

__global__ void scatter_csr_kernel(const int* __restrict__ dst, int* __restrict__ cursor,
                                   int* __restrict__ csr, int e) {
  const int i = blockIdx.x * blockDim.x + threadIdx.x;
  if (i >= e) return;
  const int pos = atomicAdd(&cursor[dst[i]], 1);
  csr[pos] = i;
}

// deterministic per-bucket order (sort edge ids ascending)
__global__ void bucket_sort_kernel(const int* __restrict__ offs, int* __restrict__ csr, int n) {
  const int v = blockIdx.x * blockDim.x + threadIdx.x;
  if (v >= n) return;
  const int b = offs[v], e = offs[v + 1];
  for (int i = b + 1; i < e; i++) {
    int key = csr[i]; int j = i - 1;
    while (j >= b && csr[j] > key) { csr[j + 1] = csr[j]; j--; }
    csr[j + 1] = key;
  }
}

// ========== genconv softmax-aggregation (wave per node, x_ln L2-resident) ==========
__global__ void genconv_aggr_kernel(const float* __restrict__ xln,
                                    const float* __restrict__ x,
                                    const int* __restrict__ src,
                                    const int* __restrict__ offs,
                                    const int* __restrict__ csr,
                                    const float* __restrict__ tptr,
                                    float* __restrict__ out, int n) {
  const int wv = threadIdx.x >> 5, lane = threadIdx.x & 31;
  const int v = blockIdx.x * 8 + wv;
  if (v >= n) return;
  const float t = tptr[0];
  const int b = offs[v], e = offs[v + 1];
  for (int dc = 0; dc < 4; dc++) {
    const int d = dc * 32 + lane;
    float mx = -INFINITY;
    for (int j = b; j < e; j++) {
      const int u = src[csr[j]];
      float m = xln[(size_t)u * 128 + d];
      m = (m > 0.f ? m : 0.f) + 1e-7f;
      mx = fmaxf(mx, m * t);
    }
    if (!(mx > -INFINITY && mx < INFINITY)) mx = 0.f;   // where(isfinite, smax, 0)
    float den = 0.f, num = 0.f;
    for (int j = b; j < e; j++) {
      const int u = src[csr[j]];
      float m = xln[(size_t)u * 128 + d];
      m = (m > 0.f ? m : 0.f) + 1e-7f;
      const float ex = expf(m * t - mx);
      den += ex; num += m * ex;
    }
    out[(size_t)v * 128 + d] = num / (den + 1e-16f) + x[(size_t)v * 128 + d];
  }
}

// ================= global-scalar stats (genconv LN over all elements) =================
__global__ void reduce_stats_kernel(const float* __restrict__ x, size_t n,
                                    float* __restrict__ p1, float* __restrict__ p2) {
  __shared__ float s1[256], s2[256];
  float a = 0.f, b = 0.f;
  for (size_t i = (size_t)blockIdx.x * 256 + threadIdx.x; i < n; i += (size_t)gridDim.x * 256) {
    const float v = x[i]; a += v; b += v * v;
  }
  s1[threadIdx.x] = a; s2[threadIdx.x] = b; __syncthreads();
  for (int o = 128; o > 0; o >>= 1) {
    if ((int)threadIdx.x < o) { s1[threadIdx.x] += s1[threadIdx.x + o]; s2[threadIdx.x] += s2[threadIdx.x + o]; }
    __syncthreads();
  }
  if (threadIdx.x == 0) { p1[blockIdx.x] = s1[0]; p2[blockIdx.x] = s2[0]; }
}

__global__ void finalize_stats_kernel(const float* __restrict__ p1, const float* __restrict__ p2,
                                      int m, float invn, float* __restrict__ stats) {
  __shared__ float s1[1024], s2[1024];
  const int t = threadIdx.x;
  s1[t] = (t < m) ? p1[t] : 0.f;
  s2[t] = (t < m) ? p2[t] : 0.f;
  __syncthreads();
  for (int o = 512; o > 0; o >>= 1) {
    if (t < o) { s1[t] += s1[t + o]; s2[t] += s2[t + o]; }
    __syncthreads();
  }
  if (t == 0) {
    const float mu = s1[0] * invn;
    const float var = s2[0] * invn - mu * mu;
    stats[0] = mu;
    stats[1] = rsqrtf(var + LN_EPS);
  }
}

__global__ void norm_relu_kernel(float* __restrict__ h, const float* __restrict__ stats,
                                 const float* __restrict__ w, const float* __restrict__ b,
                                 size_t n, int cmask) {
  const size_t i = (size_t)blockIdx.x * blockDim.x + threadIdx.x;
  if (i >= n) return;
  const int c = (int)(i & (size_t)cmask);
  float v = (h[i] - stats[0]) * stats[1] * w[c] + b[c];
  h[i] = v > 0.f ? v : 0.f;
}

// ================= threefry2x32 noise + sort keys =================
__device__ __forceinline__ unsigned rotl32(unsigned x, int r) {
  return (x << r) | (x >> (32 - r));
}
__global__ void make_keys_kernel(const int* __restrict__ degS,
                                 float* __restrict__ keys, int* __restrict__ vals, int n) {
  const int i = blockIdx.x * blockDim.x + threadIdx.x;
  const int half = n / 2;
  if (i >= half) return;
  const unsigned ks0 = 0u, ks1 = 42u, ks2 = 0u ^ 42u ^ 0x1BD11BDAu;
  unsigned x0 = (unsigned)i, x1 = (unsigned)(i + half);
  x0 += ks0; x1 += ks1;
  const int R0[4] = {13, 15, 26, 6};
  const int R1[4] = {17, 29, 16, 24};
#pragma unroll
  for (int g = 0; g < 5; g++) {
    const int* R = (g & 1) ? R1 : R0;
#pragma unroll
    for (int j = 0; j < 4; j++) { x0 += x1; x1 = rotl32(x1, R[j]); x1 ^= x0; }
    if (g == 0) { x0 += ks1; x1 += ks2 + 1u; }
    else if (g == 1) { x0 += ks2; x1 += ks0 + 2u; }
    else if (g == 2) { x0 += ks0; x1 += ks1 + 3u; }
    else if (g == 3) { x0 += ks1; x1 += ks2 + 4u; }
    else { x0 += ks2; x1 += ks0 + 5u; }
  }
  const float u0 = __uint_as_float((x0 >> 9) | 0x3f800000u) - 1.0f;
  const float u1 = __uint_as_float((x1 >> 9) | 0x3f800000u) - 1.0f;
  keys[i] = (float)degS[i] + u0 * 0.01f;
  vals[i] = i;
  keys[i + half] = (float)degS[i + half] + u1 * 0.01f;
  vals[i + half] = i + half;
}

__global__ void bitonic_kernel(float* __restrict__ keys, int* __restrict__ vals,
                               int j, int k, int n) {
  const int i = blockIdx.x * blockDim.x + threadIdx.x;
  if (i >= n) return;
  const int ixj = i ^ j;
  if (ixj <= i) return;
  const bool up = ((i & k) == 0);
  const float a = keys[i], b = keys[ixj];
  if ((up && a > b) || (!up && a < b)) {
    keys[i] = b; keys[ixj] = a;
    const int t = vals[i]; vals[i] = vals[ixj]; vals[ixj] = t;
  }
}

__global__ void inverse_perm_kernel(const int* __restrict__ order, int* __restrict__ rev, int n) {
  const int i = blockIdx.x * blockDim.x + threadIdx.x;
  if (i < n) rev[order[i]] = i;
}

// ================= depthwise conv (+SiLU) =================
__global__ void dwconv_silu_kernel(const float* __restrict__ in, int ldin,
                                   const float* __restrict__ w,
                                   const float* __restrict__ b,
                                   float* __restrict__ out, int L, int padl) {
  const int idx = blockIdx.x * blockDim.x + threadIdx.x;
  if (idx >= L * DD) return;
  const int l = idx / DD, d = idx - l * DD;
  float acc = b[d];
#pragma unroll
  for (int k = 0; k < 4; k++) {
    const int t = l + k - padl;
    if (t >= 0 && t < L) acc += in[(size_t)t * ldin + d] * w[d * 4 + k];
  }
  out[(size_t)l * DD + d] = siluf(acc);
}

// ================= mamba chunked scan =================
__global__ void scan_phase1_kernel(const float* __restrict__ dt,
                                   const float* __restrict__ xc,
                                   const float* __restrict__ proj,
                                   const float* __restrict__ Alog,
                                   float* __restrict__ Pb, float* __restrict__ Qb) {
  const int gid = blockIdx.x * blockDim.x + threadIdx.x;  // chunk*2048 + series
  const int chunk = gid >> 11;
  const int series = gid & 2047;
  const int d = series >> 4, s = series & 15;
  const float Aval = -expf(Alog[d * 16 + s]);
  float P = 1.f, Q = 0.f;
  const int t0 = chunk * CH;
  for (int t = t0; t < t0 + CH; t++) {
    const float dtv = dt[(size_t)t * 128 + d];
    const float a = expf(dtv * Aval);
    const float b = dtv * xc[(size_t)t * 128 + d] * proj[(size_t)t * PROJW + DTRANK + s];
    Q = a * Q + b;
    P *= a;
  }
  Pb[gid] = P; Qb[gid] = Q;
}

__global__ void scan_phase2_kernel(const float* __restrict__ Pb, const float* __restrict__ Qb,
                                   float* __restrict__ hinit) {
  const int series = blockIdx.x * blockDim.x + threadIdx.x;  // 2048
  float carry = 0.f;
  for (int c = 0; c < NCH; c++) {
    hinit[c * 2048 + series] = carry;
    carry = Pb[c * 2048 + series] * carry + Qb[c * 2048 + series];
  }
}

__global__ void scan_phase3_kernel(const float* __restrict__ dt,
                                   const float* __restrict__ xc,
                                   const float* __restrict__ proj,
                                   const float* __restrict__ Alog,
                                   const float* __restrict__ hinit,
                                   float* __restrict__ ys) {
  // grid: NCH * 8 blocks of 256 threads (16 channels x 16 states)
  const int chunk = blockIdx.x >> 3;
  const int dg = blockIdx.x & 7;
  const int d = dg * 16 + (threadIdx.x >> 4);
  const int s = threadIdx.x & 15;
  const float Aval = -expf(Alog[d * 16 + s]);
  float h = hinit[chunk * 2048 + d * 16 + s];
  const int t0 = chunk * CH;
  for (int t = t0; t < t0 + CH; t++) {
    const float dtv = dt[(size_t)t * 128 + d];
    const float a = expf(dtv * Aval);
    const float b = dtv * xc[(size_t)t * 128 + d] * proj[(size_t)t * PROJW + DTRANK + s];
    h = a * h + b;
    float contrib = h * proj[(size_t)t * PROJW + DTRANK + DSTATE + s];
    contrib += __shfl_xor(contrib, 1, 16);
    contrib += __shfl_xor(contrib, 2, 16);
    contrib += __shfl_xor(contrib, 4, 16);
    contrib += __shfl_xor(contrib, 8, 16);
    if (s == 0) ys[(size_t)t * 128 + d] = contrib;
  }
}

// ================= elementwise fusions =================
__global__ void ymix_kernel(float* __restrict__ ys, const float* __restrict__ xc,
                            const float* __restrict__ xz, const float* __restrict__ mD,
                            int n) {
  const int i = blockIdx.x * blockDim.x + threadIdx.x;
  if (i >= n) return;
  const int c = i & 127;
  const int row = i >> 7;
  const float z = xz[(size_t)row * 256 + 128 + c];
  ys[i] = (ys[i] + xc[i] * mD[c]) * siluf(z);
}

__global__ void mul_silu_kernel(float* __restrict__ a, const float* __restrict__ b, int n) {
  const int i = blockIdx.x * blockDim.x + threadIdx.x;
  if (i < n) a[i] *= siluf(b[i]);
}

__global__ void add_kernel(float* __restrict__ a, const float* __restrict__ b, int n) {
  const int i = blockIdx.x * blockDim.x + threadIdx.x;
  if (i < n) a[i] += b[i];
}

// ================= batchnorm stats (deterministic two-stage) =================
__global__ void bn_partial_kernel(const float* __restrict__ x,
                                  float* __restrict__ p1, float* __restrict__ p2) {
  const int c = threadIdx.x;              // 128 columns
  const int chunk = blockIdx.x;           // 128 chunks of 256 rows
  const int r0 = chunk * 256;
  float s = 0.f, q = 0.f;
  for (int r = 0; r < 256; r++) {
    const float v = x[(size_t)(r0 + r) * 128 + c];
    s += v; q += v * v;
  }
  p1[chunk * 128 + c] = s;
  p2[chunk * 128 + c] = q;
}

__global__ void bn_finalize_kernel(const float* __restrict__ p1, const float* __restrict__ p2,
                                   float* __restrict__ mu, float* __restrict__ istd) {
  const int c = threadIdx.x;
  float s = 0.f, q = 0.f;
  for (int k = 0; k < 128; k++) { s += p1[k * 128 + c]; q += p2[k * 128 + c]; }
  const float m = s * (1.f / (float)NN);
  const float var = q * (1.f / (float)NN) - m * m;
  mu[c] = m;
  istd[c] = rsqrtf(var + BN_EPS);
}

// ====================================================================
extern "C" void kernel_launch(void* const* d_in, const int* in_sizes, int n_in,
                              void* d_out, int out_size, void* d_ws, size_t ws_size,
                              hipStream_t stream) {
  (void)in_sizes; (void)n_in; (void)out_size; (void)ws_size;
  // ------- inputs (setup_inputs order) -------
  const float* x        = (const float*)d_in[0];
  const int*   ei       = (const int*)d_in[1];
  const int*   src      = ei;
  const int*   dst      = ei + EE;
  const float* ln1_w    = (const float*)d_in[2];
  const float* ln1_b    = (const float*)d_in[3];
  const float* gen_w1   = (const float*)d_in[4];
  const float* gen_b1   = (const float*)d_in[5];
  const float* gen_lnw  = (const float*)d_in[6];
  const float* gen_lnb  = (const float*)d_in[7];
  const float* gen_w2   = (const float*)d_in[8];
  const float* gen_b2   = (const float*)d_in[9];
  const float* gen_t    = (const float*)d_in[10];
  const float* inproj_w = (const float*)d_in[11];
  const float* dwc_w    = (const float*)d_in[12];
  const float* dwc_b    = (const float*)d_in[13];
  const float* m_inproj = (const float*)d_in[14];
  const float* m_conv_w = (const float*)d_in[15];
  const float* m_conv_b = (const float*)d_in[16];
  const float* m_xproj  = (const float*)d_in[17];
  const float* m_dtw    = (const float*)d_in[18];
  const float* m_dtb    = (const float*)d_in[19];
  const float* m_Alog   = (const float*)d_in[20];
  const float* m_D      = (const float*)d_in[21];
  const float* m_outprj = (const float*)d_in[22];
  const float* outproj  = (const float*)d_in[23];
  const float* fn_w     = (const float*)d_in[24];
  const float* fn_b     = (const float*)d_in[25];
  const float* mlp_w1   = (const float*)d_in[26];
  const float* mlp_b1   = (const float*)d_in[27];
  const float* mlp_w2   = (const float*)d_in[28];
  const float* mlp_b2   = (const float*)d_in[29];
  const float* bn_w     = (const float*)d_in[30];
  const float* bn_b     = (const float*)d_in[31];
  float* out = (float*)d_out;

  // ------- workspace carve -------
  char* wsb = (char*)d_ws;
  size_t off = 0;
  auto alloc = [&](size_t bytes) -> void* {
    void* p = wsb + off;
    off += (bytes + 255) & ~(size_t)255;
    return p;
  };
  const size_t ND = (size_t)NN * DD;
  float* x_ln  = (float*)alloc(ND * 4);            // also reused as hmid later
  float* g_a   = (float*)alloc(ND * 4);
  float* big   = (float*)alloc((size_t)NN * 512 * 4); // genconv hidden (N x 256) / MLP hidden (N x 512)
  float* x_loc = (float*)alloc(ND * 4);
  float* xp    = (float*)alloc(ND * 4);
  float* xs    = (float*)alloc(ND * 4);
  float* xz    = (float*)alloc((size_t)NN * 256 * 4);
  float* xc    = (float*)alloc(ND * 4);
  float* proj  = (float*)alloc((size_t)NN * PROJW * 4);
  float* dtb_  = (float*)alloc(ND * 4);
  float* ys    = (float*)alloc(ND * 4);
  float* my    = (float*)alloc(ND * 4);
  float* x_glo = (float*)alloc(ND * 4);
  float* Pb    = (float*)alloc((size_t)NCH * 2048 * 4);
  float* Qb    = (float*)alloc((size_t)NCH * 2048 * 4);
  float* hinit = (float*)alloc((size_t)NCH * 2048 * 4);
  float* part1 = (float*)alloc(1024 * 4);
  float* part2 = (float*)alloc(1024 * 4);
  float* stats = (float*)alloc(2 * 4);
  float* bnp1  = (float*)alloc(128 * 128 * 4);
  float* bnp2  = (float*)alloc(128 * 128 * 4);
  float* bnmu  = (float*)alloc(128 * 4);
  float* bnistd= (float*)alloc(128 * 4);
  float* keys  = (float*)alloc((size_t)NN * 4);
  int* vals    = (int*)alloc((size_t)NN * 4);     // == order
  int* rev     = (int*)alloc((size_t)NN * 4);
  int* degS    = (int*)alloc((size_t)NN * 4);
  int* degI    = (int*)alloc((size_t)NN * 4);
  int* offs    = (int*)alloc((size_t)(NN + 1) * 4);
  int* cursor  = (int*)alloc((size_t)NN * 4);
  int* csr     = (int*)alloc((size_t)EE * 4);
  float* hmid  = x_ln;   // reuse after genconv aggregation is done

  // ------- 1. x_ln = layernorm(x) -------
  row_ln_kernel<<<NN, 128, 0, stream>>>(x, ln1_w, ln1_b, x_ln);

  // ------- 2. degrees + CSR (deterministic bucket order) -------
  hipMemsetAsync(degS, 0, (size_t)NN * 4, stream);
  hipMemsetAsync(degI, 0, (size_t)NN * 4, stream);
  degree_kernel<<<(EE + 255) / 256, 256, 0, stream>>>(src, dst, degS, degI, EE);
  prefix_kernel<<<1, 1024, 0, stream>>>(degI, offs, NN);
  copy_int_kernel<<<(NN + 255) / 256, 256, 0, stream>>>(offs, cursor, NN);
  scatter_csr_kernel<<<(EE + 255) / 256, 256, 0, stream>>>(dst, cursor, csr, EE);
  bucket_sort_kernel<<<(NN + 255) / 256, 256, 0, stream>>>(offs, csr, NN);

  // ------- 3. genconv aggregation: g_a = aggr + x -------
  genconv_aggr_kernel<<<(NN + 7) / 8, 256, 0, stream>>>(x_ln, x, src, offs, csr, gen_t, g_a, NN);

  // ------- 4. h1 = g_a @ gen_w1 + b1 ; global-scalar LN ; relu ; @ gen_w2 + b2 -------
  {
    const int tiles = (NN / 16) * (256 / 16);
    gemm_wmma_f32<<<(tiles + 3) / 4, 128, 0, stream>>>(g_a, DD, gen_w1, gen_b1,
                                                       nullptr, nullptr, nullptr,
                                                       big, NN, DD, 256, 0);
  }
  {
    const size_t n = (size_t)NN * 256;
    reduce_stats_kernel<<<1024, 256, 0, stream>>>(big, n, part1, part2);
    finalize_stats_kernel<<<1, 1024, 0, stream>>>(part1, part2, 1024, 1.f / (float)n, stats);
    norm_relu_kernel<<<(int)((n + 255) / 256), 256, 0, stream>>>(big, stats, gen_lnw, gen_lnb, n, 255);
  }
  {
    const int tiles = (NN / 16) * (DD / 16);
    gemm_wmma_f32<<<(tiles + 3) / 4, 128, 0, stream>>>(big, 256, gen_w2, gen_b2,
                                                       nullptr, nullptr, nullptr,
                                                       x_loc, NN, 256, DD, 0);
  }

  // ------- 5. order = argsort(deg_src + threefry_noise) ; rev -------
  make_keys_kernel<<<(NN / 2 + 255) / 256, 256, 0, stream>>>(degS, keys, vals, NN);
  for (int k = 2; k <= NN; k <<= 1)
    for (int j = k >> 1; j > 0; j >>= 1)
      bitonic_kernel<<<(NN + 255) / 256, 256, 0, stream>>>(keys, vals, j, k, NN);
  inverse_perm_kernel<<<(NN + 255) / 256, 256, 0, stream>>>(vals, rev, NN);

  // ------- 6. xs = silu(dwconv(x[order] @ inproj_w, pad (1,2))) -------
  {
    const int tiles = (NN / 16) * (DD / 16);
    gemm_wmma_f32<<<(tiles + 3) / 4, 128, 0, stream>>>(x, DD, inproj_w, nullptr,
                                                       nullptr, vals, nullptr,
                                                       xp, NN, DD, DD, 0);
  }
  dwconv_silu_kernel<<<(NN * DD + 255) / 256, 256, 0, stream>>>(xp, DD, dwc_w, dwc_b, xs, NN, 1);

  // ------- 7. mamba -------
  {
    const int tiles = (NN / 16) * (256 / 16);
    gemm_wmma_f32<<<(tiles + 3) / 4, 128, 0, stream>>>(xs, DD, m_inproj, nullptr,
                                                       nullptr, nullptr, nullptr,
                                                       xz, NN, DD, 256, 0);
  }
  dwconv_silu_kernel<<<(NN * DD + 255) / 256, 256, 0, stream>>>(xz, 256, m_conv_w, m_conv_b, xc, NN, 3);
  {
    const int tiles = (NN / 16) * ((PROJW + 15) / 16);
    gemm_wmma_f32<<<(tiles + 3) / 4, 128, 0, stream>>>(xc, DD, m_xproj, nullptr,
                                                       nullptr, nullptr, nullptr,
                                                       proj, NN, DD, PROJW, 0);
  }
  {
    const int tiles = (NN / 16) * (DD / 16);
    gemm_wmma_f32<<<(tiles + 3) / 4, 128, 0, stream>>>(proj, PROJW, m_dtw, m_dtb,
                                                       nullptr, nullptr, nullptr,
                                                       dtb_, NN, DTRANK, DD, 3 /*softplus*/);
  }
  scan_phase1_kernel<<<(NCH * 2048) / 256, 256, 0, stream>>>(dtb_, xc, proj, m_Alog, Pb, Qb);
  scan_phase2_kernel<<<2048 / 256, 256, 0, stream>>>(Pb, Qb, hinit);
  scan_phase3_kernel<<<NCH * 8, 256, 0, stream>>>(dtb_, xc, proj, m_Alog, hinit, ys);
  ymix_kernel<<<(NN * DD + 255) / 256, 256, 0, stream>>>(ys, xc, xz, m_D, NN * DD);
  {
    const int tiles = (NN / 16) * (DD / 16);
    gemm_wmma_f32<<<(tiles + 3) / 4, 128, 0, stream>>>(ys, DD, m_outprj, nullptr,
                                                       nullptr, nullptr, nullptr,
                                                       my, NN, DD, DD, 0);
  }

  // ------- 8. x_glo = ((my * silu(xs)) @ outproj_w) scattered back -------
  mul_silu_kernel<<<(NN * DD + 255) / 256, 256, 0, stream>>>(my, xs, NN * DD);
  {
    const int tiles = (NN / 16) * (DD / 16);
    gemm_wmma_f32<<<(tiles + 3) / 4, 128, 0, stream>>>(my, DD, outproj, nullptr,
                                                       nullptr, nullptr, vals /*scatter*/,
                                                       x_glo, NN, DD, DD, 0);
  }

  // ------- 9. fuse + BN + LN -------
  add_kernel<<<(NN * DD + 255) / 256, 256, 0, stream>>>(x_glo, x_loc, NN * DD);
  bn_partial_kernel<<<128, 128, 0, stream>>>(x_glo, bnp1, bnp2);
  bn_finalize_kernel<<<1, 128, 0, stream>>>(bnp1, bnp2, bnmu, bnistd);
  bn_ln_kernel<<<NN, 128, 0, stream>>>(x_glo, bnmu, bnistd, bn_w, bn_b, fn_w, fn_b, hmid);

  // ------- 10. MLP + residual -------
  {
    const int tiles = (NN / 16) * (512 / 16);
    gemm_wmma_f32<<<(tiles + 3) / 4, 128, 0, stream>>>(hmid, DD, mlp_w1, mlp_b1,
                                                       nullptr, nullptr, nullptr,
                                                       big, NN, DD, 512, 2 /*gelu exact*/);
  }
  {
    const int tiles = (NN / 16) * (DD / 16);
    gemm_wmma_f32<<<(tiles + 3) / 4, 128, 0, stream>>>(big, 512, mlp_w2, mlp_b2,
                                                       x /*residual*/, nullptr, nullptr,
                                                       out, NN, 512, DD, 0);
  }
}